// PFNLayer_24386824307259
// MI455X (gfx1250) — compile-verified
//
#include <hip/hip_runtime.h>
#include <stdint.h>
#include <stddef.h>

// ---------------------------------------------------------------------------
// PFNLayer preprocessing for MI455X (gfx1250).
// Bandwidth-bound (~340 MB traffic -> ~15us @ 23.3 TB/s). No matrix math, so
// no WMMA; instead use the CDNA5 async-tensor path: double-buffered
// global_load_async_to_lds_b128 staging (ASYNCcnt), wave32 shuffle reduction,
// and non-temporal streaming stores.
// ---------------------------------------------------------------------------

#define WAVES_PER_BLOCK 8
#define THREADS (WAVES_PER_BLOCK * 32)

#define X_VOXEL 0.16f
#define X_OFFSET 0.08f   // voxel/2 + range0 ; reference bug: also used for y

__global__ __launch_bounds__(256) void pfn_coors_copy_kernel(
    const int* __restrict__ coors, float* __restrict__ out, int n)
{
    int i = blockIdx.x * blockDim.x + threadIdx.x;
    if (i < n) {
        __builtin_nontemporal_store((float)coors[i], out + i);
    }
}

__global__ __launch_bounds__(THREADS) void pfn_features_kernel(
    const float* __restrict__ pillars,      // (P, 32, 4)
    const int*   __restrict__ coors,        // (P, 4)
    const int*   __restrict__ npts,         // (P,)
    float*       __restrict__ outF,         // (P, 9, 32)
    int P, int totalWaves)
{
    // per-wave double buffer: 2 x 32 lanes x float4 = 1 KB/wave, 8 KB/block
    __shared__ __align__(16) float4 smem[WAVES_PER_BLOCK * 2 * 32];

    const int lane        = threadIdx.x & 31;
    const int waveInBlock = threadIdx.x >> 5;
    const int waveId      = blockIdx.x * WAVES_PER_BLOCK + waveInBlock;

    float4* wbuf = &smem[waveInBlock * 64];
    // raw LDS byte offsets of this lane's 16B slot in each buffer
    const unsigned lds0 = (unsigned)(uintptr_t)(wbuf + lane);
    const unsigned lds1 = (unsigned)(uintptr_t)(wbuf + 32 + lane);

    int p = waveId;
    if (p >= P) return;

    // prologue: stage pillar p into buffer 0 (16B per lane, 512B per pillar)
    {
        uint64_t ga = (uint64_t)(uintptr_t)(pillars + ((size_t)p * 32 + lane) * 4);
        asm volatile("global_load_async_to_lds_b128 %0, %1, off"
                     :: "v"(lds0), "v"(ga) : "memory");
    }

    int cur = 0;
    for (; p < P; p += totalWaves) {
        // issue next pillar's async DMA into the other buffer (branchless:
        // clamp on the last iteration -> harmless redundant load)
        int pn  = p + totalWaves;
        int pnc = (pn < P) ? pn : p;
        {
            unsigned dst = (cur == 0) ? lds1 : lds0;
            uint64_t ga  = (uint64_t)(uintptr_t)(pillars + ((size_t)pnc * 32 + lane) * 4);
            asm volatile("global_load_async_to_lds_b128 %0, %1, off"
                         :: "v"(dst), "v"(ga) : "memory");
        }
        // wait for the older buffer only; the just-issued DMA stays in flight
        asm volatile("s_wait_asynccnt 0x1" ::: "memory");

        float4 v = wbuf[cur * 32 + lane];   // ds_load_b128

        // wave32 sum over all 32 point slots (padding included, per reference)
        float sx = v.x, sy = v.y, sz = v.z;
        #pragma unroll
        for (int o = 16; o > 0; o >>= 1) {
            sx += __shfl_xor(sx, o, 32);
            sy += __shfl_xor(sy, o, 32);
            sz += __shfl_xor(sz, o, 32);
        }

        int   cnt = npts[p];
        float inv = 1.0f / (float)cnt;
        float mx = sx * inv, my = sy * inv, mz = sz * inv;

        float cxs = (float)coors[(size_t)p * 4 + 1] * X_VOXEL + X_OFFSET;
        float cys = (float)coors[(size_t)p * 4 + 2] * X_VOXEL + X_OFFSET; // ref bug: x_offset
        float offx = v.x - cxs;
        float offy = v.y - cys;

        float m = (lane < cnt) ? 1.0f : 0.0f;

        // (P, 9, 32): channel-major rows; each store = 128B coalesced line
        float* fb = outF + (size_t)p * 288 + lane;
        __builtin_nontemporal_store(offx * m,       fb);
        __builtin_nontemporal_store(offy * m,       fb + 32);
        __builtin_nontemporal_store(v.z * m,        fb + 64);
        __builtin_nontemporal_store(v.w * m,        fb + 96);
        __builtin_nontemporal_store((v.x - mx) * m, fb + 128);
        __builtin_nontemporal_store((v.y - my) * m, fb + 160);
        __builtin_nontemporal_store((v.z - mz) * m, fb + 192);
        __builtin_nontemporal_store(offx * m,       fb + 224);
        __builtin_nontemporal_store(offy * m,       fb + 256);

        cur ^= 1;
    }
}

extern "C" void kernel_launch(void* const* d_in, const int* in_sizes, int n_in,
                              void* d_out, int out_size, void* d_ws, size_t ws_size,
                              hipStream_t stream) {
    const float* pillars = (const float*)d_in[0];  // (P, 32, 4) f32
    const int*   coors   = (const int*)d_in[1];    // (P, 4) i32
    const int*   npts    = (const int*)d_in[2];    // (P,)  i32

    const int P  = in_sizes[2];
    const int nC = in_sizes[1];                    // P*4 coor elements
    float* out = (float*)d_out;

    // output tuple: [coors as float (P*4)] ++ [features (P*9*32)]
    int cb = (nC + 255) / 256;
    hipLaunchKernelGGL(pfn_coors_copy_kernel, dim3(cb), dim3(256), 0, stream,
                       coors, out, nC);

    int needBlocks = (P + WAVES_PER_BLOCK - 1) / WAVES_PER_BLOCK;
    int blocks = needBlocks < 2500 ? needBlocks : 2500;  // persistent waves
    int totalWaves = blocks * WAVES_PER_BLOCK;
    hipLaunchKernelGGL(pfn_features_kernel, dim3(blocks), dim3(THREADS), 0, stream,
                       pillars, coors, npts, out + nC, P, totalWaves);
}